// SentenceEncoder_batch_52725018525818
// MI455X (gfx1250) — compile-verified
//
#include <hip/hip_runtime.h>
#include <hip/hip_bf16.h>
#include <math.h>

#define B_   64
#define N_   512
#define E_   128
#define F_   128
#define PAD_ 49999
#define LBL_ 2
#define SST  (N_ + 4)   // padded LDS row stride: lane m -> bank (4m+k)%64, conflict-free

typedef float v2f __attribute__((ext_vector_type(2)));
typedef float v8f __attribute__((ext_vector_type(8)));

__device__ __forceinline__ float wave_max32(float x) {
  #pragma unroll
  for (int off = 16; off > 0; off >>= 1) x = fmaxf(x, __shfl_xor(x, off, 32));
  return x;
}
__device__ __forceinline__ float wave_sum32(float x) {
  #pragma unroll
  for (int off = 16; off > 0; off >>= 1) x += __shfl_xor(x, off, 32);
  return x;
}

// ---------------- kernel 0: zero the pool accumulator ----------------
__global__ void k_zero(float* __restrict__ p, int n) {
  int i = blockIdx.x * blockDim.x + threadIdx.x;
  if (i < n) p[i] = 0.0f;
}

// ---------------- kernel 1: h = emb[doc] @ W  (WMMA f32 16x16x4) ----------------
// grid (N/16, B), block 32 (one wave per 16-row tile of h)
__global__ void __launch_bounds__(32) k_h_gemm(const int* __restrict__ doc,
                                               const float* __restrict__ emb,
                                               const float* __restrict__ W,
                                               float* __restrict__ h) {
  const int itile = blockIdx.x;
  const int b     = blockIdx.y;
  const int lane  = threadIdx.x;
  const int mrow  = lane & 15;
  const int hi    = lane >> 4;                 // 0: K%4 in {0,1}, 1: K%4 in {2,3}
  const int tok   = doc[b * N_ + itile * 16 + mrow];
  const float* erow = emb + (long)tok * E_;

  for (int ct = 0; ct < F_ / 16; ++ct) {
    v8f c = {};
    #pragma unroll 4
    for (int kk = 0; kk < E_ / 4; ++kk) {
      const int k0 = kk * 4 + hi * 2;
      v2f a  = { erow[k0], erow[k0 + 1] };                                       // A: 16x4
      v2f bb = { W[k0 * F_ + ct * 16 + mrow], W[(k0 + 1) * F_ + ct * 16 + mrow] }; // B: 4x16
      c = __builtin_amdgcn_wmma_f32_16x16x4_f32(false, a, false, bb,
                                                (short)0, c, false, false);
    }
    #pragma unroll
    for (int r = 0; r < 8; ++r) {
      const int M = r + hi * 8;
      h[((long)(b * N_ + itile * 16 + M)) * F_ + ct * 16 + mrow] = c[r];
    }
  }
}

// ---------------- kernel 2: e_src/e_dst dot products + pad flags ----------------
__global__ void k_scores(const int* __restrict__ doc, const float* __restrict__ h,
                         const float* __restrict__ a_src, const float* __restrict__ a_dst,
                         float* __restrict__ e_src, float* __restrict__ e_dst,
                         int* __restrict__ pad) {
  const int idx = blockIdx.x * blockDim.x + threadIdx.x;   // b*N + n
  if (idx >= B_ * N_) return;
  const float* hr = h + (long)idx * F_;
  float s0 = 0.0f, s1 = 0.0f;
  #pragma unroll 4
  for (int f = 0; f < F_; ++f) {
    const float v = hr[f];
    s0 += v * a_src[f];
    s1 += v * a_dst[f];
  }
  e_src[idx] = s0;
  e_dst[idx] = s1;
  pad[idx]   = (doc[idx] == PAD_) ? 1 : 0;
}

// ---------------- kernel 3 (fused): masked softmax -> LDS + NT store,
//                  then sentence = elu(att @ h) straight from LDS ----------------
// grid (N/16, B), block 256 = 8 waves.
//   phase 1: wave w softmaxes rows 2w, 2w+1 of the 16-row i-tile into LDS (and d_out, NT)
//   phase 2: wave w computes the 16x16 output f-tile w with A from LDS, B = h from L2
__global__ void __launch_bounds__(256) k_att_sent(const float* __restrict__ e_src,
                                                  const float* __restrict__ e_dst,
                                                  const int* __restrict__ pad,
                                                  const float* __restrict__ h,
                                                  float* __restrict__ att,
                                                  float* __restrict__ sent,
                                                  float* __restrict__ pool) {
  __shared__ float satt[16 * SST];             // 16 attention rows, padded stride
  const int itile = blockIdx.x;
  const int b     = blockIdx.y;
  const int wave  = threadIdx.x >> 5;
  const int lane  = threadIdx.x & 31;

  // ---- phase 1: softmax of 2 rows per wave ----
  #pragma unroll
  for (int rr = 0; rr < 2; ++rr) {
    const int rloc = wave * 2 + rr;            // 0..15
    const int i    = itile * 16 + rloc;
    const float ei  = e_src[b * N_ + i];
    const int  padi = pad[b * N_ + i];
    float v[16];
    float m = -3.0e38f;
    #pragma unroll
    for (int t = 0; t < 16; ++t) {
      const int j = t * 32 + lane;
      const int masked = padi | pad[b * N_ + j];
      const float x = masked ? -1.0e9f : fmaxf(ei + e_dst[b * N_ + j], 0.0f); // SLOPE=0 relu
      v[t] = x;
      m = fmaxf(m, x);
    }
    m = wave_max32(m);
    float s = 0.0f;
    #pragma unroll
    for (int t = 0; t < 16; ++t) { v[t] = __expf(v[t] - m); s += v[t]; }
    s = wave_sum32(s);
    const float inv = 1.0f / s;
    float* ar = att + ((long)b * N_ + i) * N_;
    #pragma unroll
    for (int t = 0; t < 16; ++t) {
      const float r = v[t] * inv;
      satt[rloc * SST + t * 32 + lane] = r;
      __builtin_nontemporal_store(r, &ar[t * 32 + lane]);   // attention never re-read
    }
  }
  __syncthreads();

  // ---- phase 2: 16x16 WMMA f-tile per wave, K = 512 from LDS ----
  const int mrow = lane & 15;
  const int hi   = lane >> 4;
  const float* hb = h + (long)b * N_ * F_;
  const int fcol  = wave * 16 + mrow;

  v8f c = {};
  #pragma unroll 4
  for (int kk = 0; kk < N_ / 4; ++kk) {
    const int k0 = kk * 4 + hi * 2;
    if ((kk & 15) == 0 && kk + 16 < N_ / 4)
      __builtin_prefetch(&hb[(kk * 4 + 64) * F_ + fcol], 0, 1);  // global_prefetch_b8
    v2f a  = { satt[mrow * SST + k0], satt[mrow * SST + k0 + 1] };   // ds_load
    v2f bb = { hb[k0 * F_ + fcol], hb[(k0 + 1) * F_ + fcol] };
    c = __builtin_amdgcn_wmma_f32_16x16x4_f32(false, a, false, bb,
                                              (short)0, c, false, false);
  }

  float colsum = 0.0f;
  #pragma unroll
  for (int r = 0; r < 8; ++r) {
    float x = c[r];
    x = (x > 0.0f) ? x : (__expf(x) - 1.0f);   // elu, alpha=1
    __builtin_nontemporal_store(x,
        &sent[((long)(b * N_ + itile * 16 + r + hi * 8)) * F_ + fcol]);
    colsum += x;
  }
  // lanes L and L+16 hold the same output column f = wave*16 + (L&15)
  colsum += __shfl_xor(colsum, 16, 32);
  if (lane < 16) atomicAdd(&pool[b * F_ + wave * 16 + lane], colsum);
}

// ---------------- kernel 4: pool mean + classifier + softmax over batch axis ----------------
__global__ void k_final(const float* __restrict__ pool, const float* __restrict__ Wc,
                        const float* __restrict__ bc, float* __restrict__ pool_out,
                        float* __restrict__ label_out) {
  const int t = threadIdx.x;                 // 128 threads
  __shared__ float lg[B_ * LBL_];
  const float invN = 1.0f / (float)N_;
  for (int idx = t; idx < B_ * F_; idx += 128) pool_out[idx] = pool[idx] * invN;

  const int b = t >> 1, l = t & 1;
  float acc = bc[l];
  for (int f = 0; f < F_; ++f) acc += pool[b * F_ + f] * invN * Wc[f * LBL_ + l];
  lg[t] = acc;
  __syncthreads();

  float m = -3.0e38f;
  for (int bb = 0; bb < B_; ++bb) m = fmaxf(m, lg[bb * LBL_ + l]);
  float s = 0.0f;
  for (int bb = 0; bb < B_; ++bb) s += __expf(lg[bb * LBL_ + l] - m);
  label_out[t] = __expf(lg[t] - m) / s;
}

extern "C" void kernel_launch(void* const* d_in, const int* in_sizes, int n_in,
                              void* d_out, int out_size, void* d_ws, size_t ws_size,
                              hipStream_t stream) {
  const int*   doc   = (const int*)d_in[0];
  const float* emb   = (const float*)d_in[1];
  const float* W     = (const float*)d_in[2];
  const float* a_src = (const float*)d_in[3];
  const float* a_dst = (const float*)d_in[4];
  const float* Wc    = (const float*)d_in[5];
  const float* bc    = (const float*)d_in[6];

  float* out       = (float*)d_out;
  float* pool_out  = out;                                    // (B, F)
  float* att_out   = out + (size_t)B_ * F_;                  // (B, N, N)
  float* sent_out  = att_out + (size_t)B_ * N_ * N_;         // (B, N, F)
  float* label_out = sent_out + (size_t)B_ * N_ * F_;        // (B, LBL)

  float* h     = (float*)d_ws;                               // B*N*F floats (16 MB)
  float* e_src = h + (size_t)B_ * N_ * F_;
  float* e_dst = e_src + (size_t)B_ * N_;
  int*   pad   = (int*)(e_dst + (size_t)B_ * N_);
  float* pool  = (float*)(pad + (size_t)B_ * N_);            // B*F accumulator

  k_zero<<<(B_ * F_ + 255) / 256, 256, 0, stream>>>(pool, B_ * F_);
  k_h_gemm<<<dim3(N_ / 16, B_), 32, 0, stream>>>(doc, emb, W, h);
  k_scores<<<(B_ * N_ + 255) / 256, 256, 0, stream>>>(doc, h, a_src, a_dst, e_src, e_dst, pad);
  k_att_sent<<<dim3(N_ / 16, B_), 256, 0, stream>>>(e_src, e_dst, pad, h, att_out, sent_out, pool);
  k_final<<<1, 128, 0, stream>>>(pool, Wc, bc, pool_out, label_out);
}